// RegimeSwitchingKF_74448963109262
// MI455X (gfx1250) — compile-verified
//
#include <hip/hip_runtime.h>
#include <cstdint>
#include <cstddef>

#define N_SERIES 4096
#define NTIME    400
#define O        9
#define LDIM     4
#define BLOCK    256
#define LOG2PI_F 1.8378770664093453f
#define LN2_F    0.6931471805599453f
#define INVLN2_F 1.4426950408889634f

// ---- detect gfx1250 async global->LDS path ----------------------------------
#if defined(__has_builtin)
#if __has_builtin(__builtin_amdgcn_global_load_async_to_lds_b32) && \
    __has_builtin(__builtin_amdgcn_s_wait_asynccnt)
#define USE_ASYNC_LDS 1
#endif
#endif
#ifndef USE_ASYNC_LDS
#define USE_ASYNC_LDS 0
#endif

#define AS_GLOBAL __attribute__((address_space(1)))
#define AS_LDS    __attribute__((address_space(3)))

#if USE_ASYNC_LDS
// Issue O async b32 copies global->LDS; imm offset must be a compile-time
// constant, so unroll via template recursion.
template <int K>
__device__ __forceinline__ void async_copy_row(const float* g, float* l) {
  __builtin_amdgcn_global_load_async_to_lds_b32(
      (AS_GLOBAL int*)g, (AS_LDS int*)l, 4 * K, 0);
  if constexpr (K + 1 < O) async_copy_row<K + 1>(g, l);
}
#endif

// ---- fast hardware math (v_rsq/v_rcp/v_log/v_exp) ---------------------------
__device__ __forceinline__ float f_rsqrt(float x) {
  float r = __builtin_amdgcn_rsqf(x);
  return r * (1.5f - 0.5f * x * r * r);   // one NR step
}
__device__ __forceinline__ float f_rcp(float x) {
  float r = __builtin_amdgcn_rcpf(x);
  return r * (2.0f - x * r);              // one NR step
}
__device__ __forceinline__ float f_log(float x) { return __builtin_amdgcn_logf(x) * LN2_F; }
__device__ __forceinline__ float f_exp(float x) { return __builtin_amdgcn_exp2f(x * INVLN2_F); }

// packed symmetric index helpers (compile-time after unrolling)
__device__ __forceinline__ constexpr int SIDX(int i, int j) {
  return (i >= j) ? (i * (i + 1) / 2 + j) : (j * (j + 1) / 2 + i);
}
#define TIDX(i, j) ((i) * ((i) + 1) / 2 + (j))  // requires i >= j

// Both loading matrices have exactly one nonzero per row: column c[k], weight w[k].
struct Reg1 { static constexpr int c[O] = {0, 0, 0, 1, 1, 1, 2, 2, 2}; };
struct Reg2 { static constexpr int c[O] = {3, 3, 3, 3, 3, 3, 3, 3, 3}; };

// One regime's Kalman update + innovation log-likelihood, all in registers.
template <typename RC>
__device__ __forceinline__ float regime_update(
    const float w[O], const float yv[O], const float ep[LDIM],
    const float Pp[10], const float Rj[O], const float Rd[O],
    float eta_u[LDIM], float Pu[10]) {
  // innovation v = y - L eta_pred
  float v[O];
#pragma unroll
  for (int k = 0; k < O; ++k) v[k] = yv[k] - w[k] * ep[RC::c[k]];

  // F = L Pp L^T + R + jitter I  (packed lower, 45 entries)
  float F[45];
#pragma unroll
  for (int i = 0; i < O; ++i) {
#pragma unroll
    for (int j = 0; j <= i; ++j) {
      float f = w[i] * w[j] * Pp[SIDX(RC::c[i], RC::c[j])];
      if (i == j) f += Rj[i];
      F[TIDX(i, j)] = f;
    }
  }

  // in-place Cholesky, keep 1/L_kk and sum(log d) = 2 sum(log L_kk)
  float rinv[O];
  float sumlogd = 0.f;
#pragma unroll
  for (int k = 0; k < O; ++k) {
    float d = F[TIDX(k, k)];
#pragma unroll
    for (int j = 0; j < k; ++j) { float l = F[TIDX(k, j)]; d -= l * l; }
    float ri = f_rsqrt(d);
    rinv[k] = ri;
    sumlogd += f_log(d);                 // log L_kk = 0.5 log d
    F[TIDX(k, k)] = d * ri;
#pragma unroll
    for (int i = k + 1; i < O; ++i) {
      float s = F[TIDX(i, k)];
#pragma unroll
      for (int j = 0; j < k; ++j) s -= F[TIDX(i, j)] * F[TIDX(k, j)];
      F[TIDX(i, k)] = s * ri;
    }
  }

  // z = c \ v ; loglik
  float z[O];
#pragma unroll
  for (int i = 0; i < O; ++i) {
    float s = v[i];
#pragma unroll
    for (int j = 0; j < i; ++j) s -= F[TIDX(i, j)] * z[j];
    z[i] = s * rinv[i];
  }
  float q = 0.f;
#pragma unroll
  for (int k = 0; k < O; ++k) q += z[k] * z[k];
  float loglik = -4.5f * LOG2PI_F - 0.5f * sumlogd - 0.5f * q;

  // X = F^{-1} (L Pp) : forward + backward triangular solves, 4 RHS.
  // LP[i][col] = w[i] * Pp[c[i], col]; K[l][o] = X[o][l]
  float X[O][LDIM];
#pragma unroll
  for (int i = 0; i < O; ++i) {
#pragma unroll
    for (int col = 0; col < LDIM; ++col) {
      float s = w[i] * Pp[SIDX(RC::c[i], col)];
#pragma unroll
      for (int j = 0; j < i; ++j) s -= F[TIDX(i, j)] * X[j][col];
      X[i][col] = s * rinv[i];
    }
  }
#pragma unroll
  for (int i = O - 1; i >= 0; --i) {
#pragma unroll
    for (int col = 0; col < LDIM; ++col) {
      float s = X[i][col];
#pragma unroll
      for (int j = i + 1; j < O; ++j) s -= F[TIDX(j, i)] * X[j][col];
      X[i][col] = s * rinv[i];
    }
  }

  // eta_u = ep + K v
#pragma unroll
  for (int l = 0; l < LDIM; ++l) {
    float s = ep[l];
#pragma unroll
    for (int o = 0; o < O; ++o) s += X[o][l] * v[o];
    eta_u[l] = s;
  }

  // A = I - K L   (sparse L: contributes only where c[o] == j)
  float A[LDIM][LDIM];
#pragma unroll
  for (int l = 0; l < LDIM; ++l) {
#pragma unroll
    for (int j = 0; j < LDIM; ++j) {
      float s = (l == j) ? 1.f : 0.f;
#pragma unroll
      for (int o = 0; o < O; ++o)
        if (RC::c[o] == j) s -= X[o][l] * w[o];   // constexpr-folds
      A[l][j] = s;
    }
  }
  // Pu = A Pp A^T + K diag(R) K^T  (Joseph form, symmetric)
  float M[LDIM][LDIM];
#pragma unroll
  for (int l = 0; l < LDIM; ++l)
#pragma unroll
    for (int m = 0; m < LDIM; ++m) {
      float s = 0.f;
#pragma unroll
      for (int j = 0; j < LDIM; ++j) s += A[l][j] * Pp[SIDX(j, m)];
      M[l][m] = s;
    }
#pragma unroll
  for (int l = 0; l < LDIM; ++l)
#pragma unroll
    for (int m = 0; m <= l; ++m) {
      float s = 0.f;
#pragma unroll
      for (int j = 0; j < LDIM; ++j) s += M[l][j] * A[m][j];
#pragma unroll
      for (int o = 0; o < O; ++o) s += X[o][l] * Rd[o] * X[o][m];
      Pu[TIDX(l, m)] = s;
    }
  return loglik;
}

__global__ __launch_bounds__(BLOCK) void imm_kf_kernel(
    const float* __restrict__ y, const float* __restrict__ B1,
    const float* __restrict__ B2, const float* __restrict__ l1f,
    const float* __restrict__ l2f, const float* __restrict__ lq,
    const float* __restrict__ lr, const float* __restrict__ g0,
    const float* __restrict__ gc, float* __restrict__ probs_out,
    float* __restrict__ ypred_out, float* __restrict__ ll_partial) {
  const int tid = threadIdx.x;
  const int n = blockIdx.x * BLOCK + tid;

  // ---- uniform model constants (scalar loads -> SGPRs) ----
  float b1[3][3];
#pragma unroll
  for (int i = 0; i < 3; ++i)
#pragma unroll
    for (int j = 0; j < 3; ++j) b1[i][j] = B1[i * 3 + j];
  const float b2 = B2[0];
  float w1[O] = {1.f, l1f[0], l1f[1], 1.f, l1f[2], l1f[3], 1.f, l1f[4], l1f[5]};
  float w2[O];
  w2[0] = 1.f;
#pragma unroll
  for (int k = 0; k < 8; ++k) w2[k + 1] = l2f[k];
  float Qd[LDIM];
#pragma unroll
  for (int i = 0; i < LDIM; ++i) Qd[i] = f_exp(lq[i]);
  float Rd[O], Rj[O];
#pragma unroll
  for (int k = 0; k < O; ++k) { Rd[k] = f_exp(lr[k]); Rj[k] = Rd[k] + 1e-6f; }
  const float gi = g0[0];
  const float gc0 = gc[0], gc1 = gc[1], gc2 = gc[2];

  // ---- per-series filter state, all registers ----
  float prob1 = 0.99f, prob2 = 0.01f;
  float eta[LDIM] = {0.f, 0.f, 0.f, 0.f};
  float P[10];
#pragma unroll
  for (int l = 0; l < LDIM; ++l)
#pragma unroll
    for (int m = 0; m <= l; ++m) P[TIDX(l, m)] = (l == m) ? 1000.f : 0.f;
  float ll_acc = 0.f;

  const float* gser = y + (size_t)n * (NTIME * O);

#if USE_ASYNC_LDS
  __shared__ float ybuf[2][BLOCK * O];
  // prime the pipeline: async-copy t=0 row into buffer 0
  async_copy_row<0>(gser, &ybuf[0][tid * O]);
  int buf = 0;
#endif

#pragma unroll 1
  for (int t = 0; t < NTIME; ++t) {
    float yv[O];
#if USE_ASYNC_LDS
    __builtin_amdgcn_s_wait_asynccnt(0);
#pragma unroll
    for (int k = 0; k < O; ++k) yv[k] = ybuf[buf][tid * O + k];
    if (t + 1 < NTIME) {  // prefetch next step into the other buffer
      async_copy_row<0>(gser + (size_t)(t + 1) * O, &ybuf[buf ^ 1][tid * O]);
    }
    buf ^= 1;
#else
    const float* gcur = gser + (size_t)t * O;
    if (t + 1 < NTIME) __builtin_prefetch(gcur + O, 0, 3);  // global_prefetch_b8
#pragma unroll
    for (int k = 0; k < O; ++k) yv[k] = gcur[k];
#endif

    // regime-stay probability p11 = sigmoid(g0 + eta[:3] . gamma)
    float sarg = gi + gc0 * eta[0] + gc1 * eta[1] + gc2 * eta[2];
    float p11 = f_rcp(1.f + f_exp(-sarg));

    // predict: eta_pred = B eta ; P_pred = B P B^T + Q (block-diag B)
    float ep[LDIM];
#pragma unroll
    for (int i = 0; i < 3; ++i)
      ep[i] = b1[i][0] * eta[0] + b1[i][1] * eta[1] + b1[i][2] * eta[2];
    ep[3] = b2 * eta[3];
    float T[LDIM][LDIM];
#pragma unroll
    for (int i = 0; i < 3; ++i)
#pragma unroll
      for (int j = 0; j < LDIM; ++j)
        T[i][j] = b1[i][0] * P[SIDX(0, j)] + b1[i][1] * P[SIDX(1, j)] +
                  b1[i][2] * P[SIDX(2, j)];
#pragma unroll
    for (int j = 0; j < LDIM; ++j) T[3][j] = b2 * P[SIDX(3, j)];
    float Pp[10];
#pragma unroll
    for (int i = 0; i < LDIM; ++i)
#pragma unroll
      for (int j = 0; j <= i; ++j) {
        float s = (j < 3) ? (T[i][0] * b1[j][0] + T[i][1] * b1[j][1] +
                             T[i][2] * b1[j][2])
                          : (T[i][3] * b2);
        if (i == j) s += Qd[i];
        Pp[TIDX(i, j)] = s;
      }

    // two independent regime updates (ILP pair)
    float eta1[LDIM], P1[10], eta2[LDIM], P2[10];
    float ll1 = regime_update<Reg1>(w1, yv, ep, Pp, Rj, Rd, eta1, P1);
    float ll2 = regime_update<Reg2>(w2, yv, ep, Pp, Rj, Rd, eta2, P2);

    // IMM mixing (p21 = 0, p22 = 1)
    float num1 = f_exp(ll1) * (prob1 * p11);
    float num2 = f_exp(ll2) * (prob1 * (1.f - p11) + prob2);
    float marg = num1 + num2 + 1e-9f;
    float im = f_rcp(marg);
    float pr1 = num1 * im, pr2 = num2 * im;
    ll_acc += f_log(marg);

#pragma unroll
    for (int l = 0; l < LDIM; ++l) eta[l] = pr1 * eta1[l] + pr2 * eta2[l];
#pragma unroll
    for (int l = 0; l < LDIM; ++l)
#pragma unroll
      for (int m = 0; m <= l; ++m) {
        float d1 = (eta1[l] - eta[l]) * (eta1[m] - eta[m]);
        float d2 = (eta2[l] - eta[l]) * (eta2[m] - eta[m]);
        P[TIDX(l, m)] = pr1 * (P1[TIDX(l, m)] + d1) + pr2 * (P2[TIDX(l, m)] + d2);
      }
    prob1 = pr1;
    prob2 = pr2;

    // outputs: filtered probs + predicted y (contiguous per series over t -> L2-friendly)
    const size_t base = (size_t)n * NTIME + (size_t)t;
    probs_out[base * 2 + 0] = pr1;
    probs_out[base * 2 + 1] = pr2;
#pragma unroll
    for (int k = 0; k < O; ++k)
      ypred_out[base * O + k] =
          pr1 * w1[k] * eta[Reg1::c[k]] + pr2 * w2[k] * eta[Reg2::c[k]];
  }

  // deterministic block tree-reduction of log-likelihood
  __shared__ float red[BLOCK];
  red[tid] = ll_acc;
  __syncthreads();
#pragma unroll
  for (int s = BLOCK / 2; s > 0; s >>= 1) {
    if (tid < s) red[tid] += red[tid + s];
    __syncthreads();
  }
  if (tid == 0) ll_partial[blockIdx.x] = red[0];
}

__global__ void ll_reduce_kernel(const float* __restrict__ part,
                                 float* __restrict__ out, int nparts) {
  if (blockIdx.x == 0 && threadIdx.x == 0) {
    float s = 0.f;
    for (int i = 0; i < nparts; ++i) s += part[i];   // fixed order: deterministic
    out[0] = -s;
  }
}

extern "C" void kernel_launch(void* const* d_in, const int* in_sizes, int n_in,
                              void* d_out, int out_size, void* d_ws,
                              size_t ws_size, hipStream_t stream) {
  (void)in_sizes; (void)n_in; (void)out_size; (void)ws_size;
  const float* y   = (const float*)d_in[0];
  const float* B1  = (const float*)d_in[1];
  const float* B2  = (const float*)d_in[2];
  const float* l1f = (const float*)d_in[3];
  const float* l2f = (const float*)d_in[4];
  const float* lq  = (const float*)d_in[5];
  const float* lr  = (const float*)d_in[6];
  const float* g0  = (const float*)d_in[7];
  const float* gc  = (const float*)d_in[8];

  float* out = (float*)d_out;
  float* probs = out + 1;                                   // [N, NT, 2]
  float* ypred = out + 1 + (size_t)N_SERIES * NTIME * 2;    // [N, NT, 9]
  float* llpart = (float*)d_ws;                             // 16 block partials

  const int nblocks = N_SERIES / BLOCK;  // 16
  imm_kf_kernel<<<dim3(nblocks), dim3(BLOCK), 0, stream>>>(
      y, B1, B2, l1f, l2f, lq, lr, g0, gc, probs, ypred, llpart);
  ll_reduce_kernel<<<dim3(1), dim3(32), 0, stream>>>(llpart, out, nblocks);
}